// PYTORCH_FS_77524159693458
// MI455X (gfx1250) — compile-verified
//
#include <hip/hip_runtime.h>
#include <hip/hip_bf16.h>

typedef float v2f __attribute__((ext_vector_type(2)));
typedef float v8f __attribute__((ext_vector_type(8)));

#define EPSF 1e-5f

// ---------------------------------------------------------------------------
// Kernel 1: camera projection via V_WMMA_F32_16X16X4_F32.
// One wave (32 lanes) handles one tile of 16 queries for one camera.
//   A (16x4, MxK): lanes 0-15 -> (K0,K1)=(px,py); lanes 16-31 -> (K2,K3)=(pz,1)
//   B (4x16, KxN): N = output row i of lidar2img; B[k][n] = L[n][k]; n>=4 -> 0
//   D (16x16):     D[m][n] = cam coordinate i=n of query m
// D is scattered through LDS so lane m gathers its query's (x,y,z), then the
// perspective divide / normalization / frustum mask are computed per lane.
// ---------------------------------------------------------------------------
__global__ __launch_bounds__(32)
void proj_wmma_kernel(const float* __restrict__ ref,       // [nq,3]
                      const float* __restrict__ pc_range,  // [6]
                      const float* __restrict__ img_shape, // [2] = (H,W)
                      const float* __restrict__ l2i,       // [num_cam,4,4]
                      float* __restrict__ ws_grid,         // [num_cam,nq,2]
                      float* __restrict__ out_mask,        // [nq,num_cam]
                      int nq, int num_cam, int ntiles)
{
    const int t    = blockIdx.x;
    const int cam  = t / ntiles;
    const int tile = t - cam * ntiles;
    const int lane = threadIdx.x;
    const int half = lane >> 4;     // K-half selector
    const int m    = lane & 15;     // row (query within tile) / col (cam row)
    const int q    = tile * 16 + m;

    const float r0 = pc_range[0], r1 = pc_range[1], r2 = pc_range[2];
    const float s0 = pc_range[3] - r0;
    const float s1 = pc_range[4] - r1;
    const float s2 = pc_range[5] - r2;

    float px = 0.f, py = 0.f, pz = 0.f;
    if (q < nq) {
        px = ref[q * 3 + 0] * s0 + r0;
        py = ref[q * 3 + 1] * s1 + r1;
        pz = ref[q * 3 + 2] * s2 + r2;
    }

    v2f a;
    if (half == 0) { a.x = px; a.y = py; }
    else           { a.x = pz; a.y = 1.0f; }

    const float* __restrict__ L = l2i + (size_t)cam * 16;
    v2f b; b.x = 0.f; b.y = 0.f;
    if (m < 4) {
        if (half == 0) { b.x = L[m * 4 + 0]; b.y = L[m * 4 + 1]; }
        else           { b.x = L[m * 4 + 2]; b.y = L[m * 4 + 3]; }
    }

    v8f cacc = {};
    // D = A x B + 0   (emits v_wmma_f32_16x16x4_f32)
    v8f d = __builtin_amdgcn_wmma_f32_16x16x4_f32(
        /*neg_a=*/false, a, /*neg_b=*/false, b,
        /*c_mod=*/(short)0, cacc, /*reuse_a=*/false, /*reuse_b=*/false);

    // Scatter D (cols 0..3 only) to LDS: lds[query_row][cam_row]
    __shared__ float lds[16 * 4];
    if (m < 4) {
        const int base_m = half * 8;            // lanes 16-31 hold M=8..15
        #pragma unroll
        for (int v = 0; v < 8; ++v)
            lds[(base_m + v) * 4 + m] = d[v];
    }
    __syncthreads();

    if (lane < 16 && q < nq) {
        const float cx = lds[m * 4 + 0];
        const float cy = lds[m * 4 + 1];
        const float cz = lds[m * 4 + 2];

        bool  msk = cz > EPSF;
        const float dz = fmaxf(cz, EPSF);
        float x = (cx / dz) / img_shape[1];     // divide by image W
        float y = (cy / dz) / img_shape[0];     // divide by image H
        x = (x - 0.5f) * 2.0f;
        y = (y - 0.5f) * 2.0f;
        msk = msk && (x > -1.0f) && (x < 1.0f) && (y > -1.0f) && (y < 1.0f);

        ws_grid[((size_t)cam * nq + q) * 2 + 0] = x;
        ws_grid[((size_t)cam * nq + q) * 2 + 1] = y;
        out_mask[(size_t)q * num_cam + cam] = msk ? 1.0f : 0.0f;
    }
}

// ---------------------------------------------------------------------------
// Kernel 2: bilinear gather over 4 pyramid levels.
// One 256-thread block per (query, cam); lane = channel. Feature pyramid
// (189 MB) fits in the 192 MB L2, so the stride-H*W channel gathers are
// L2-resident after first touch.
// ---------------------------------------------------------------------------
__global__ __launch_bounds__(256)
void sample_kernel(const float* __restrict__ f0, const float* __restrict__ f1,
                   const float* __restrict__ f2, const float* __restrict__ f3,
                   const float* __restrict__ ws_grid,   // [num_cam,nq,2]
                   float* __restrict__ out_feat,        // [C,nq,num_cam,4]
                   int nq, int num_cam, int C)
{
    const int p   = blockIdx.x;
    const int cam = p % num_cam;
    const int q   = p / num_cam;
    const int c   = threadIdx.x;

    const float x = ws_grid[((size_t)cam * nq + q) * 2 + 0];
    const float y = ws_grid[((size_t)cam * nq + q) * 2 + 1];

    const float* const feats[4] = { f0, f1, f2, f3 };
    const int Hs[4] = { 116, 58, 29, 15 };
    const int Ws[4] = { 200, 100, 50, 25 };

    float* __restrict__ outp =
        out_feat + (((size_t)c * nq + q) * num_cam + cam) * 4;

    #pragma unroll
    for (int l = 0; l < 4; ++l) {
        const int H = Hs[l], W = Ws[l];
        const float ix = ((x + 1.0f) * (float)W - 1.0f) * 0.5f;
        const float iy = ((y + 1.0f) * (float)H - 1.0f) * 0.5f;
        const float fx0 = floorf(ix), fy0 = floorf(iy);
        const float wx1 = ix - fx0,   wy1 = iy - fy0;
        const float wx0 = 1.0f - wx1, wy0 = 1.0f - wy1;

        const float* __restrict__ base =
            feats[l] + ((size_t)cam * C + c) * (size_t)(H * W);

        const float fys[2] = { fy0, fy0 + 1.0f };
        const float fxs[2] = { fx0, fx0 + 1.0f };
        const float wys[2] = { wy0, wy1 };
        const float wxs[2] = { wx0, wx1 };

        float acc = 0.0f;
        #pragma unroll
        for (int cy = 0; cy < 2; ++cy) {
            #pragma unroll
            for (int cx = 0; cx < 2; ++cx) {
                const float fy = fys[cy], fx = fxs[cx];
                const bool valid = (fx >= 0.0f) && (fx <= (float)(W - 1)) &&
                                   (fy >= 0.0f) && (fy <= (float)(H - 1));
                float v = 0.0f;
                if (valid) {
                    const int xi = (int)fx;
                    const int yi = (int)fy;
                    v = base[yi * W + xi];
                }
                acc += wys[cy] * wxs[cx] * v;
            }
        }
        outp[l] = acc;
    }
}

// ---------------------------------------------------------------------------
// Kernel 3: pass-through copy of reference_points into output slot 0.
// ---------------------------------------------------------------------------
__global__ __launch_bounds__(256)
void copy_kernel(const float* __restrict__ src, float* __restrict__ dst, int n)
{
    const int i = blockIdx.x * blockDim.x + threadIdx.x;
    if (i < n) dst[i] = src[i];
}

extern "C" void kernel_launch(void* const* d_in, const int* in_sizes, int n_in,
                              void* d_out, int out_size, void* d_ws, size_t ws_size,
                              hipStream_t stream)
{
    const float* f0  = (const float*)d_in[0];   // [1,6,256,116,200]
    const float* f1  = (const float*)d_in[1];   // [1,6,256, 58,100]
    const float* f2  = (const float*)d_in[2];   // [1,6,256, 29, 50]
    const float* f3  = (const float*)d_in[3];   // [1,6,256, 15, 25]
    const float* ref = (const float*)d_in[4];   // [1,900,3]
    const float* pc  = (const float*)d_in[5];   // [6]
    const float* img = (const float*)d_in[6];   // [2]
    const float* l2i = (const float*)d_in[7];   // [1,6,4,4]

    const int nq      = in_sizes[4] / 3;                      // 900
    const int num_cam = in_sizes[7] / 16;                     // 6
    const int C       = in_sizes[0] / (num_cam * 116 * 200);  // 256

    float* out       = (float*)d_out;
    const int ref_sz = in_sizes[4];                                    // 2700
    float* out_feat  = out + ref_sz;
    float* out_mask  = out + ref_sz + (size_t)C * nq * num_cam * 4;

    float* ws_grid = (float*)d_ws;      // num_cam*nq*2 floats = 43.2 KB

    const int ntiles = (nq + 15) / 16;  // 57

    proj_wmma_kernel<<<num_cam * ntiles, 32, 0, stream>>>(
        ref, pc, img, l2i, ws_grid, out_mask, nq, num_cam, ntiles);

    sample_kernel<<<nq * num_cam, C, 0, stream>>>(
        f0, f1, f2, f3, ws_grid, out_feat, nq, num_cam, C);

    copy_kernel<<<(ref_sz + 255) / 256, 256, 0, stream>>>(ref, out, ref_sz);
}